// Raindrop_47897475285618
// MI455X (gfx1250) — compile-verified
//
#include <hip/hip_runtime.h>

// Raindrop effect on MI455X (gfx1250).
//
// Roofline: 96 GFLOP fp32 total; the 157MB image fits in the 192MB L2 ->
// L2-bandwidth bound. 51-tap separable conv expressed as a banded fp32
// matmul on V_WMMA_F32_16X16X4_F32 (17 chained K=4 WMMAs per 16x16 tile,
// fp32 accumulate == reference precision).
//
// This revision removes the divergent-EXEC band lookups (zero-padded LDS
// band table -> unconditional ds_load_b32 with immediate offsets) and adds
// an interior fast path so 36/40 tiles have no reflect math and get
// compile-time-constant load offsets.

#define KMAXT 51
#define HALFW 25
#define IH 640
#define IW 640
#define NBC 96          // B*C = 32*3
#define NPIX (IH * IW)
#define BPAD 15         // band table logical offset (covers index >= -15)
#define BLEN 112        // 15 pad + 51 taps + tail pad (index <= 83)

typedef __attribute__((ext_vector_type(2))) float v2f;
typedef __attribute__((ext_vector_type(8))) float v8f;

__device__ __forceinline__ int reflect_idx(int i, int n) {
    i = (i < 0) ? -i : i;
    i = (i >= n) ? (2 * n - 2 - i) : i;
    return i;
}

// ---- per-drop 51-tap gaussian (matches gaussian_kernel1d) ----
__global__ void weights_kernel(const float* __restrict__ blur_radius, int step,
                               float* __restrict__ wk) {
    __shared__ float w[KMAXT];
    __shared__ float ssum;
    int t = threadIdx.x; // 64 threads
    float sigma = blur_radius[step];
    float ks = floorf(2.0f * sigma) + 1.0f;
    if (fmodf(ks, 2.0f) == 0.0f) ks += 1.0f;
    float half = (ks - 1.0f) * 0.5f;
    if (t < KMAXT) {
        float x = (float)t - (float)HALFW;
        float v = expf(-0.5f * (x / sigma) * (x / sigma));
        v *= (fabsf(x) <= half) ? 1.0f : 0.0f;
        w[t] = v;
    }
    __syncthreads();
    if (t == 0) {
        float s = 0.0f;
        for (int i = 0; i < KMAXT; ++i) s += w[i];
        ssum = s;
    }
    __syncthreads();
    if (t < KMAXT) wk[t] = w[t] / ssum;
}

// ---- per-drop elliptical mask over HxW (shared across all 96 channels) ----
__global__ void mask_kernel(const float* __restrict__ positions,
                            const float* __restrict__ radius, int step,
                            float* __restrict__ mask) {
    int idx = blockIdx.x * blockDim.x + threadIdx.x;
    if (idx >= NPIX) return;
    int y = idx / IW, x = idx % IW;
    float px = fminf(fmaxf(positions[2 * step + 0], -1.0f), 1.0f);
    float py = fminf(fmaxf(positions[2 * step + 1], -1.0f), 1.0f);
    float r  = fminf(fmaxf(radius[step], 40.0f), 50.0f);
    float x0 = (px + 1.0f) * 0.5f * (float)IW;
    float y0 = (py + 1.0f) * 0.5f * (float)IH;
    float wr = r, hr = r * 1.2f;
    float dy = ((float)y - y0) / hr;
    float dx = ((float)x - x0) / wr;
    float d = dy * dy + dx * dx;
    float m = expf(-powf(d, 1.8f) + 1e-10f);
    mask[idx] = fminf(fmaxf(m, 0.0f), 1.0f);
}

// Load zero-padded band table into LDS: band[BPAD + t] = k1d[t], zeros outside.
__device__ __forceinline__ void load_band(const float* __restrict__ wk,
                                          float* band) {
    int tid = threadIdx.x;
    for (int i = tid; i < BLEN; i += blockDim.x) band[i] = 0.0f;
    __syncthreads();
    if (tid < KMAXT) band[BPAD + tid] = wk[tid];
    __syncthreads();
}

// ---- vertical 51-tap conv of (img*mask), reflect pad, via fp32 WMMA ----
// D[m][n] = sum_j k1d[j-m] * (img*mask)[ybase-25+j][xbase+n],  j in [0,68)
__global__ __launch_bounds__(128) void
vconv_kernel(const float* __restrict__ img, const float* __restrict__ mask,
             const float* __restrict__ wk, float* __restrict__ tmp) {
    __shared__ float band[BLEN];
    load_band(wk, band);

    int tid  = threadIdx.x;
    int lane = tid & 31;
    int wave = tid >> 5;
    int xbase = (blockIdx.x * 4 + wave) * 16;
    int ybase = blockIdx.y * 16;
    int base  = blockIdx.z * NPIX;

    int mrow = lane & 15;                 // A: M index
    int ncol = lane & 15;                 // B: N index
    int koff = (lane < 16) ? 0 : 2;       // K split across lane halves
    int col  = xbase + ncol;
    int aidx = BPAD + koff - mrow;        // band index base for A

    v8f acc = {};
    if (ybase >= HALFW && ybase - HALFW + 67 < IH) {
        // interior: no reflection; constant per-step address offsets
        const float* ip = img  + base + (ybase - HALFW + koff) * IW + col;
        const float* mp = mask +        (ybase - HALFW + koff) * IW + col;
#pragma unroll
        for (int kk = 0; kk < 17; ++kk) {
            int o = kk * 4 * IW;
            v2f a, b;
            a.x = band[aidx + 4 * kk];
            a.y = band[aidx + 4 * kk + 1];
            b.x = ip[o] * mp[o];
            b.y = ip[o + IW] * mp[o + IW];
            acc = __builtin_amdgcn_wmma_f32_16x16x4_f32(false, a, false, b,
                                                        (short)0, acc, false, false);
        }
    } else {
        // border: reflect row indices (uniform branch, EXEC stays full)
#pragma unroll
        for (int kk = 0; kk < 17; ++kk) {
            int j0 = kk * 4;
            v2f a, b;
            a.x = band[aidx + j0];
            a.y = band[aidx + j0 + 1];
            int y0r = reflect_idx(ybase - HALFW + j0 + koff, IH);
            int y1r = reflect_idx(ybase - HALFW + j0 + koff + 1, IH);
            b.x = img[base + y0r * IW + col] * mask[y0r * IW + col];
            b.y = img[base + y1r * IW + col] * mask[y1r * IW + col];
            acc = __builtin_amdgcn_wmma_f32_16x16x4_f32(false, a, false, b,
                                                        (short)0, acc, false, false);
        }
    }
    int m0 = (lane < 16) ? 0 : 8;
#pragma unroll
    for (int i = 0; i < 8; ++i)
        tmp[base + (ybase + m0 + i) * IW + col] = acc[i];
}

// ---- horizontal 51-tap conv of tmp + mask blend (in-place safe) ----
// D[m][n] = sum_j tmp[ybase+m][xbase-25+j] * k1d[j-n]
__global__ __launch_bounds__(128) void
hconv_blend_kernel(const float* __restrict__ tmp, const float* __restrict__ mask,
                   const float* __restrict__ wk, const float* __restrict__ cur,
                   float* __restrict__ out) {
    __shared__ float band[BLEN];
    load_band(wk, band);

    int tid  = threadIdx.x;
    int lane = tid & 31;
    int wave = tid >> 5;
    int xbase = (blockIdx.x * 4 + wave) * 16;
    int ybase = blockIdx.y * 16;
    int base  = blockIdx.z * NPIX;

    int mrow = lane & 15;
    int ncol = lane & 15;
    int koff = (lane < 16) ? 0 : 2;
    int col  = xbase + ncol;
    int bidx = BPAD + koff - ncol;        // band index base for B

    v8f acc = {};
    if (xbase >= HALFW && xbase - HALFW + 67 < IW) {
        const float* tp = tmp + base + (ybase + mrow) * IW + (xbase - HALFW + koff);
#pragma unroll
        for (int kk = 0; kk < 17; ++kk) {
            int o = kk * 4;
            v2f a, b;
            a.x = tp[o];
            a.y = tp[o + 1];
            b.x = band[bidx + o];
            b.y = band[bidx + o + 1];
            acc = __builtin_amdgcn_wmma_f32_16x16x4_f32(false, a, false, b,
                                                        (short)0, acc, false, false);
        }
    } else {
        int rowoff = base + (ybase + mrow) * IW;
#pragma unroll
        for (int kk = 0; kk < 17; ++kk) {
            int j0 = kk * 4;
            v2f a, b;
            int c0 = reflect_idx(xbase - HALFW + j0 + koff, IW);
            int c1 = reflect_idx(xbase - HALFW + j0 + koff + 1, IW);
            a.x = tmp[rowoff + c0];
            a.y = tmp[rowoff + c1];
            b.x = band[bidx + j0];
            b.y = band[bidx + j0 + 1];
            acc = __builtin_amdgcn_wmma_f32_16x16x4_f32(false, a, false, b,
                                                        (short)0, acc, false, false);
        }
    }
    int m0 = (lane < 16) ? 0 : 8;
#pragma unroll
    for (int i = 0; i < 8; ++i) {
        int y = ybase + m0 + i;
        int p = base + y * IW + col;
        float mk = mask[y * IW + col];
        out[p] = acc[i] * mk + cur[p] * (1.0f - mk);
    }
}

extern "C" void kernel_launch(void* const* d_in, const int* in_sizes, int n_in,
                              void* d_out, int out_size, void* d_ws, size_t ws_size,
                              hipStream_t stream) {
    const float* img  = (const float*)d_in[0];  // 32*3*640*640
    const float* pos  = (const float*)d_in[1];  // 12*2
    const float* rad  = (const float*)d_in[2];  // 12
    const float* blur = (const float*)d_in[3];  // 12
    float* out = (float*)d_out;

    // workspace layout: taps (64f) | mask (H*W f) | tmp image (B*C*H*W f)
    char* ws = (char*)d_ws;
    float* wk   = (float*)ws;
    float* mask = (float*)(ws + 256);
    float* tmp  = (float*)(ws + 256 + (size_t)NPIX * sizeof(float));

    dim3 cgrid(IW / 16 / 4, IH / 16, NBC);  // (10, 40, 96), 4 waves/block
    dim3 cblk(128);
    int maskBlocks = (NPIX + 255) / 256;

    for (int s = 0; s < 12; ++s) {
        weights_kernel<<<1, 64, 0, stream>>>(blur, s, wk);
        mask_kernel<<<maskBlocks, 256, 0, stream>>>(pos, rad, s, mask);
        const float* cur = (s == 0) ? img : out;
        vconv_kernel<<<cgrid, cblk, 0, stream>>>(cur, mask, wk, tmp);
        hconv_blend_kernel<<<cgrid, cblk, 0, stream>>>(tmp, mask, wk, cur, out);
    }
}